// InterMaskEncoder_13365938225809
// MI455X (gfx1250) — compile-verified
//
#include <hip/hip_runtime.h>

// ---------------------------------------------------------------------------
// GIN forward (eval): 3 layers of  x <- ReLU(BN( ReLU((x+agg) W1 + b1) W2 + b2 ))
// Shapes: B=16, N=1024, F=256, E=16384, L=3.  Output: [L, B, N, F] fp32.
// fp32 WMMA (16x16x4) GEMMs, async-to-LDS double-buffered tile pipeline.
// ---------------------------------------------------------------------------

typedef __attribute__((ext_vector_type(2))) float v2f;
typedef __attribute__((ext_vector_type(8))) float v8f;

#define GIN_F      256
#define GIN_N      1024
#define GIN_B      16
#define GIN_E      16384
#define GIN_L      3
#define GIN_M      (GIN_B * GIN_N)   // 16384 rows per GEMM

#define MB 128          // rows per block (8 waves x 16)
#define NB 32           // cols per block (2 x 16 per wave)
#define KB 32           // K chunk through LDS
#define LDA (KB + 2)    // 34-float stride: 8B-aligned even offsets, conflict-free
#define LDW (KB + 2)

#define BN_EPS 1e-5f

// ---------------------------------------------------------------------------
// Scatter-add: agg[b, dst[e], f] += x[b, src[e], f]   (agg pre-initialized = x)
// ---------------------------------------------------------------------------
__global__ __launch_bounds__(GIN_F)
void gin_scatter_kernel(const float* __restrict__ x,
                        const int* __restrict__ src,
                        const int* __restrict__ dst,
                        float* __restrict__ agg) {
    const int e = blockIdx.x;
    const int b = blockIdx.y;
    const int f = threadIdx.x;
    const int s = src[e];
    const int d = dst[e];
    const float v = x[((size_t)b * GIN_N + s) * GIN_F + f];
    atomicAdd(&agg[((size_t)b * GIN_N + d) * GIN_F + f], v);
}

// ---------------------------------------------------------------------------
// Async global->LDS helpers (CDNA5, tracked by ASYNCcnt).
// lds  : wave-relative LDS byte address (low 32 bits of a generic LDS pointer)
// base : uniform 64-bit global base (SGPR pair), voff: per-lane byte offset
// ---------------------------------------------------------------------------
__device__ __forceinline__ unsigned lds_off(const void* p) {
    return (unsigned)(unsigned long long)(uintptr_t)p;  // LDS aperture: low 32 bits
}
__device__ __forceinline__ void async_ld_b64(unsigned lds, unsigned long long base,
                                             unsigned voff) {
    asm volatile("global_load_async_to_lds_b64 %0, %1, %2"
                 :: "v"(lds), "v"(voff), "s"(base) : "memory");
}
__device__ __forceinline__ void async_ld_b32(unsigned lds, unsigned long long base,
                                             unsigned voff) {
    asm volatile("global_load_async_to_lds_b32 %0, %1, %2"
                 :: "v"(lds), "v"(voff), "s"(base) : "memory");
}
__device__ __forceinline__ void wait_async0() {
    asm volatile("s_wait_asynccnt 0" ::: "memory");
}

// ---------------------------------------------------------------------------
// Fused GEMM: out = epilogue( A * W + bias )
//   mode 0 : out = relu(acc + bias)
//   mode 1 : out = relu((acc + bias)*inv + shift)   (BatchNorm eval)
// A:[GIN_M,F] row-major, W:[F,F] row-major. Block = 256 thr = 8 wave32,
// tile 128x32, K chunked by 32 via double-buffered async-to-LDS; per wave:
// one 16-row strip, two f32 16x16 accumulators (V_WMMA_F32_16X16X4_F32).
// ---------------------------------------------------------------------------
__global__ __launch_bounds__(256)
void gin_gemm_kernel(const float* __restrict__ A,
                     const float* __restrict__ W,
                     const float* __restrict__ bias,
                     const float* __restrict__ gamma,  // mode 1 only
                     const float* __restrict__ beta,
                     const float* __restrict__ mean,
                     const float* __restrict__ var,
                     float* __restrict__ out,
                     int mode) {
    __shared__ float As[2][MB][LDA];
    __shared__ float Wt[2][NB][LDW];   // transposed: Wt[buf][col][k]

    const int tid  = threadIdx.x;
    const int wave = tid >> 5;
    const int lane = tid & 31;
    const int rowBase = blockIdx.x * MB;
    const int colBase = blockIdx.y * NB;

    const unsigned long long Abase = (unsigned long long)(uintptr_t)A;
    const unsigned long long Wbase = (unsigned long long)(uintptr_t)W;

    // Per-thread async-fill assignments (fixed across chunks).
    // A tile: 128x32 floats = 2048 B64 chunks, 8 per thread (16 per row).
    const int ar  = tid >> 1;              // base pattern recomputed in loop
    (void)ar;
    // W tile: 32(K)x32(N) -> transposed scatter, 1024 B32, 4 per thread.

    auto issue_chunk = [&](int buf, int kk) {
        #pragma unroll
        for (int j = 0; j < 8; ++j) {
            const int linear = j * 256 + tid;
            const int r  = linear >> 4;          // 16 B64 chunks per 32-float row
            const int c2 = (linear & 15) << 1;   // float offset in row
            const unsigned l = lds_off(&As[buf][r][c2]);
            const unsigned g = (unsigned)((((unsigned)(rowBase + r) * GIN_F) + kk + c2)
                                          * sizeof(float));
            async_ld_b64(l, Abase, g);
        }
        #pragma unroll
        for (int j = 0; j < 4; ++j) {
            const int linear = j * 256 + tid;
            const int k = linear >> 5;
            const int c = linear & 31;
            const unsigned l = lds_off(&Wt[buf][c][k]);
            const unsigned g = (unsigned)((((unsigned)(kk + k) * GIN_F) + colBase + c)
                                          * sizeof(float));
            async_ld_b32(l, Wbase, g);
        }
    };

    v8f acc0 = {};
    v8f acc1 = {};

    const int halfK = (lane < 16) ? 0 : 2;   // ISA 16x4 A/B layout half-wave K split
    const int l15   = lane & 15;
    const int mrow  = wave * 16 + l15;

    issue_chunk(0, 0);

    const int NCHUNK = GIN_F / KB;           // 8
    for (int i = 0; i < NCHUNK; ++i) {
        const int buf = i & 1;
        wait_async0();        // this wave's fill of buf complete
        __syncthreads();      // everyone filled buf / done reading buf^1
        if (i + 1 < NCHUNK) issue_chunk(buf ^ 1, (i + 1) * KB);  // overlap w/ compute

        #pragma unroll
        for (int k4 = 0; k4 < KB; k4 += 4) {
            const v2f a  = *(const v2f*)&As[buf][mrow][k4 + halfK];
            const v2f b0 = *(const v2f*)&Wt[buf][l15][k4 + halfK];
            const v2f b1 = *(const v2f*)&Wt[buf][16 + l15][k4 + halfK];
            acc0 = __builtin_amdgcn_wmma_f32_16x16x4_f32(
                false, a, false, b0, (short)0, acc0, false, false);
            acc1 = __builtin_amdgcn_wmma_f32_16x16x4_f32(
                false, a, false, b1, (short)0, acc1, false, false);
        }
        __syncthreads();      // compute done before buf is refilled next round
    }

    // --- epilogue: D layout v<->M (v | v+8), N = lane&15 --------------------
    const int colA = colBase + l15;
    const int colB = colA + 16;
    const float bA = bias[colA];
    const float bB = bias[colB];
    float invA = 1.f, shA = 0.f, invB = 1.f, shB = 0.f;
    if (mode == 1) {
        invA = gamma[colA] * rsqrtf(var[colA] + BN_EPS);
        shA  = beta[colA] - mean[colA] * invA;
        invB = gamma[colB] * rsqrtf(var[colB] + BN_EPS);
        shB  = beta[colB] - mean[colB] * invB;
    }
    const int rowHalf = (lane < 16) ? 0 : 8;
    #pragma unroll
    for (int v = 0; v < 8; ++v) {
        const size_t row = (size_t)rowBase + wave * 16 + v + rowHalf;
        float t0 = acc0[v] + bA;
        float t1 = acc1[v] + bB;
        if (mode == 1) {
            t0 = t0 * invA + shA;
            t1 = t1 * invB + shB;
        }
        out[row * GIN_F + colA] = fmaxf(t0, 0.f);
        out[row * GIN_F + colB] = fmaxf(t1, 0.f);
    }
}

// ---------------------------------------------------------------------------
extern "C" void kernel_launch(void* const* d_in, const int* in_sizes, int n_in,
                              void* d_out, int out_size, void* d_ws, size_t ws_size,
                              hipStream_t stream) {
    (void)in_sizes; (void)n_in; (void)out_size; (void)ws_size;

    const float* h     = (const float*)d_in[0];
    const int*   edges = (const int*)  d_in[1];   // [2, E]
    const float* W1    = (const float*)d_in[2];   // [L, F, F]
    const float* b1    = (const float*)d_in[3];   // [L, F]
    const float* W2    = (const float*)d_in[4];
    const float* b2    = (const float*)d_in[5];
    const float* gamma = (const float*)d_in[6];
    const float* beta  = (const float*)d_in[7];
    const float* mean  = (const float*)d_in[8];
    const float* var   = (const float*)d_in[9];
    float* out = (float*)d_out;

    const size_t BNF = (size_t)GIN_B * GIN_N * GIN_F;   // 4,194,304 elements
    float* agg = (float*)d_ws;          // 16 MB : becomes m = x + scatter(x)
    float* tmp = agg + BNF;             // 16 MB : hidden activations

    const int* src = edges;
    const int* dst = edges + GIN_E;

    const dim3 gemmGrid(GIN_M / MB, GIN_F / NB);   // 128 x 8 blocks

    for (int i = 0; i < GIN_L; ++i) {
        const float* x = (i == 0) ? h : out + (size_t)(i - 1) * BNF;

        // agg = x ; agg[b,dst,:] += x[b,src,:]   =>  agg = m
        hipMemcpyAsync(agg, x, BNF * sizeof(float), hipMemcpyDeviceToDevice, stream);
        gin_scatter_kernel<<<dim3(GIN_E, GIN_B), GIN_F, 0, stream>>>(x, src, dst, agg);

        // tmp = relu(m @ W1[i] + b1[i])
        gin_gemm_kernel<<<gemmGrid, 256, 0, stream>>>(
            agg,
            W1 + (size_t)i * GIN_F * GIN_F, b1 + (size_t)i * GIN_F,
            nullptr, nullptr, nullptr, nullptr,
            tmp, 0);

        // out[i] = relu(BN(tmp @ W2[i] + b2[i]))
        gin_gemm_kernel<<<gemmGrid, 256, 0, stream>>>(
            tmp,
            W2 + (size_t)i * GIN_F * GIN_F, b2 + (size_t)i * GIN_F,
            gamma + (size_t)i * GIN_F, beta + (size_t)i * GIN_F,
            mean + (size_t)i * GIN_F, var + (size_t)i * GIN_F,
            out + (size_t)i * BNF, 1);
    }
}